// SimpleRank_6408091206265
// MI455X (gfx1250) — compile-verified
//
#include <hip/hip_runtime.h>
#include <math.h>

typedef __attribute__((ext_vector_type(2))) float v2f;
typedef __attribute__((ext_vector_type(8))) float v8f;

#define N_NODES 100000
#define N_EDGES 3200000
#define IN_DIM  256
#define HID_DIM 256
#define OUT_DIM 128
#define TOPK    10000

// ---------------- node init: deg = 1 (self loop), sel_map = -1 ----------------
__global__ __launch_bounds__(256) void init_nodes_kernel(float* deg, int* sel_map) {
    int i = blockIdx.x * blockDim.x + threadIdx.x;
    if (i < N_NODES) { deg[i] = 1.0f; sel_map[i] = -1; }
}

// ---------------- deg[dst] += 1 over all edges ----------------
__global__ __launch_bounds__(256) void deg_scatter_kernel(const long long* __restrict__ edst,
                                                          float* __restrict__ deg) {
    int e = blockIdx.x * blockDim.x + threadIdx.x;
    if (e < N_EDGES) atomicAdd(&deg[(int)edst[e]], 1.0f);
}

// ---------------- sel_map[indices[j]] = j ----------------
__global__ __launch_bounds__(256) void selmap_kernel(const long long* __restrict__ indices,
                                                     int* __restrict__ sel_map) {
    int j = blockIdx.x * blockDim.x + threadIdx.x;
    if (j < TOPK) sel_map[(int)indices[j]] = j;
}

// ---------------- deg -> rsqrt(deg) in place (deg >= 1 always) ----------------
__global__ __launch_bounds__(256) void dinv_kernel(float* deg) {
    int i = blockIdx.x * blockDim.x + threadIdx.x;
    if (i < N_NODES) deg[i] = rsqrtf(deg[i]);
}

// ---------------- GEMM1: hw = x @ W1  [100000,256]x[256,256], fp32 WMMA ----------------
// block = 128 (4 waves). Each wave: one 16-row tile x 4 col tiles (64 cols), A reused 4x.
__global__ __launch_bounds__(128) void gemm1_kernel(const float* __restrict__ x,
                                                    const float* __restrict__ W1,
                                                    float* __restrict__ hw) {
    const int lane = threadIdx.x & 31;
    const int wave = threadIdx.x >> 5;          // 0..3
    const int m0   = blockIdx.x * 16;
    const int mr   = lane & 15;
    const int hi   = lane >> 4;                 // 0 or 1
    const int kb   = hi * 2;                    // local K base {0,2}
    const int cb   = wave * 64;

    const float* xrow = x + (size_t)(m0 + mr) * IN_DIM;
    v8f c0 = {}, c1 = {}, c2 = {}, c3 = {};

    for (int k = 0; k < IN_DIM; k += 4) {
        v2f a;
        a.x = xrow[k + kb];
        a.y = xrow[k + kb + 1];
        const float* wq = W1 + (size_t)(k + kb) * HID_DIM;   // row k+kb; next row +HID_DIM
        const int n0 = cb + mr;
        v2f b0, b1, b2, b3;
        b0.x = wq[n0     ]; b0.y = wq[HID_DIM + n0     ];
        b1.x = wq[n0 + 16]; b1.y = wq[HID_DIM + n0 + 16];
        b2.x = wq[n0 + 32]; b2.y = wq[HID_DIM + n0 + 32];
        b3.x = wq[n0 + 48]; b3.y = wq[HID_DIM + n0 + 48];
        c0 = __builtin_amdgcn_wmma_f32_16x16x4_f32(false, a, false, b0, (short)0, c0, false, false);
        c1 = __builtin_amdgcn_wmma_f32_16x16x4_f32(false, a, false, b1, (short)0, c1, false, false);
        c2 = __builtin_amdgcn_wmma_f32_16x16x4_f32(false, a, false, b2, (short)0, c2, false, false);
        c3 = __builtin_amdgcn_wmma_f32_16x16x4_f32(false, a, false, b3, (short)0, c3, false, false);
    }
    for (int j = 0; j < 8; ++j) {
        const int row = m0 + j + 8 * hi;
        float* o = hw + (size_t)row * HID_DIM + cb + mr;
        o[0]  = c0[j];
        o[16] = c1[j];
        o[32] = c2[j];
        o[48] = c3[j];
    }
}

// ---------------- agg init with self-loop term: agg[j] = hw[idx[j]] * dinv^2 ----------------
__global__ __launch_bounds__(256) void agg_init_kernel(const long long* __restrict__ indices,
                                                       const float* __restrict__ hw,
                                                       const float* __restrict__ dinv,
                                                       float* __restrict__ agg) {
    int t = blockIdx.x * blockDim.x + threadIdx.x;
    if (t >= TOPK * HID_DIM) return;
    const int j = t >> 8;
    const int d = t & 255;
    const int node = (int)indices[j];
    const float di = dinv[node];
    agg[t] = hw[(size_t)node * HID_DIM + d] * di * di;
}

// ---------------- filtered edge scatter: wave per edge, only selected dst ----------------
__global__ __launch_bounds__(256) void edge_scatter_kernel(const long long* __restrict__ esrc,
                                                           const long long* __restrict__ edst,
                                                           const int* __restrict__ sel_map,
                                                           const float* __restrict__ dinv,
                                                           const float* __restrict__ hw,
                                                           float* __restrict__ agg) {
    const int lane = threadIdx.x & 31;
    const int wave = threadIdx.x >> 5;
    const long long e = (long long)blockIdx.x * 8 + wave;
    if (e >= N_EDGES) return;
    const int dst = (int)edst[e];
    const int j = sel_map[dst];
    if (j < 0) return;                              // 90% of edges exit here (uniform per wave)
    const int src = (int)esrc[e];
    const float norm = dinv[src] * dinv[dst];
    const float4* hrow = (const float4*)(hw + (size_t)src * HID_DIM);
    float* arow = agg + (size_t)j * HID_DIM;
    const float4 p = hrow[lane * 2];
    const float4 q = hrow[lane * 2 + 1];
    const int d0 = lane * 8;
    atomicAdd(arow + d0 + 0, p.x * norm);
    atomicAdd(arow + d0 + 1, p.y * norm);
    atomicAdd(arow + d0 + 2, p.z * norm);
    atomicAdd(arow + d0 + 3, p.w * norm);
    atomicAdd(arow + d0 + 4, q.x * norm);
    atomicAdd(arow + d0 + 5, q.y * norm);
    atomicAdd(arow + d0 + 6, q.z * norm);
    atomicAdd(arow + d0 + 7, q.w * norm);
}

// ---------------- GEMM2: out = relu(agg + b1) @ W2 + b2, fp32 WMMA ----------------
// block = 256 (8 waves). Wave w computes 16x16 tile at cols [16w, 16w+16).
__global__ __launch_bounds__(256) void gemm2_kernel(const float* __restrict__ agg,
                                                    const float* __restrict__ b1,
                                                    const float* __restrict__ W2,
                                                    const float* __restrict__ b2,
                                                    float* __restrict__ out) {
    const int lane = threadIdx.x & 31;
    const int wave = threadIdx.x >> 5;          // 0..7
    const int m0   = blockIdx.x * 16;
    const int mr   = lane & 15;
    const int hi   = lane >> 4;
    const int kb   = hi * 2;
    const int n    = wave * 16 + mr;

    const float* arow = agg + (size_t)(m0 + mr) * HID_DIM;
    v8f c = {};
    for (int k = 0; k < HID_DIM; k += 4) {
        v2f a, b;
        a.x = fmaxf(arow[k + kb]     + b1[k + kb],     0.0f);
        a.y = fmaxf(arow[k + kb + 1] + b1[k + kb + 1], 0.0f);
        b.x = W2[(size_t)(k + kb)     * OUT_DIM + n];
        b.y = W2[(size_t)(k + kb + 1) * OUT_DIM + n];
        c = __builtin_amdgcn_wmma_f32_16x16x4_f32(false, a, false, b, (short)0, c, false, false);
    }
    const float bias = b2[n];
    for (int j = 0; j < 8; ++j) {
        const int row = m0 + j + 8 * hi;
        out[(size_t)row * OUT_DIM + n] = c[j] + bias;
    }
}

// ---------------- row-wise log_softmax over 128 cols, wave per row, in place ----------------
__global__ __launch_bounds__(256) void logsoftmax_kernel(float* __restrict__ out) {
    const int lane = threadIdx.x & 31;
    const int wave = threadIdx.x >> 5;
    const int row = blockIdx.x * 8 + wave;
    if (row >= TOPK) return;
    float* r = out + (size_t)row * OUT_DIM;
    float v0 = r[lane], v1 = r[lane + 32], v2 = r[lane + 64], v3 = r[lane + 96];
    float mx = fmaxf(fmaxf(v0, v1), fmaxf(v2, v3));
    for (int off = 16; off > 0; off >>= 1) mx = fmaxf(mx, __shfl_xor(mx, off, 32));
    float s = expf(v0 - mx) + expf(v1 - mx) + expf(v2 - mx) + expf(v3 - mx);
    for (int off = 16; off > 0; off >>= 1) s += __shfl_xor(s, off, 32);
    const float lg = mx + logf(s);
    r[lane]      = v0 - lg;
    r[lane + 32] = v1 - lg;
    r[lane + 64] = v2 - lg;
    r[lane + 96] = v3 - lg;
}

extern "C" void kernel_launch(void* const* d_in, const int* in_sizes, int n_in,
                              void* d_out, int out_size, void* d_ws, size_t ws_size,
                              hipStream_t stream) {
    const float*     x       = (const float*)d_in[0];
    const long long* eidx    = (const long long*)d_in[1];   // [2, N_EDGES]
    const long long* indices = (const long long*)d_in[2];   // [TOPK]
    const float*     W1      = (const float*)d_in[3];
    const float*     b1      = (const float*)d_in[4];
    const float*     W2      = (const float*)d_in[5];
    const float*     b2      = (const float*)d_in[6];
    float* out = (float*)d_out;

    const long long* esrc = eidx;
    const long long* edst = eidx + N_EDGES;

    // workspace partition (256B aligned offsets)
    char* ws = (char*)d_ws;
    float* dinv    = (float*)(ws + 0);                      //   400,000 B
    int*   sel_map = (int*)  (ws + 400128);                 //   400,000 B
    float* agg     = (float*)(ws + 800256);                 // 10,240,000 B
    float* hw      = (float*)(ws + 11040256);               // 102,400,000 B
    (void)ws_size; (void)n_in; (void)in_sizes; (void)out_size;

    init_nodes_kernel  <<<(N_NODES + 255) / 256, 256, 0, stream>>>(dinv, sel_map);
    deg_scatter_kernel <<<(N_EDGES + 255) / 256, 256, 0, stream>>>(edst, dinv);
    selmap_kernel      <<<(TOPK + 255) / 256,    256, 0, stream>>>(indices, sel_map);
    dinv_kernel        <<<(N_NODES + 255) / 256, 256, 0, stream>>>(dinv);

    gemm1_kernel       <<<N_NODES / 16, 128, 0, stream>>>(x, W1, hw);

    agg_init_kernel    <<<(TOPK * HID_DIM) / 256, 256, 0, stream>>>(indices, hw, dinv, agg);
    edge_scatter_kernel<<<N_EDGES / 8, 256, 0, stream>>>(esrc, edst, sel_map, dinv, hw, agg);

    gemm2_kernel       <<<TOPK / 16, 256, 0, stream>>>(agg, b1, W2, b2, out);
    logsoftmax_kernel  <<<(TOPK + 7) / 8, 256, 0, stream>>>(out);
}